// SeparateFC_24515673326016
// MI455X (gfx1250) — compile-verified
//
#include <hip/hip_runtime.h>

// ---------------------------------------------------------------------------
// SeparateFC: out[n,p,d] = sum_c x[n,p,c] * W[p,d,c] + b[p,d]
// N=8192, P=128, D=64, all fp32.
//
// Memory-bound (AI ~ 16 FLOP/B): floor = (268MB x + 268MB out)/23.3TB/s ~ 23us.
// Strategy: fp32 WMMA (v_wmma_f32_16x16x4_f32), x read exactly once per
// element, W_p staged in padded LDS (bank-conflict-free ds_load_b64 for B
// fragments), bias folded into the accumulator init.
// ---------------------------------------------------------------------------

typedef __attribute__((ext_vector_type(2))) float v2f;
typedef __attribute__((ext_vector_type(8))) float v8f;

#define N_TOT 8192
#define P_TOT 128
#define D_DIM 64
#define LDS_STRIDE 68   // 64 + 4 pad: lane->lane delta = 68 floats, 68%64=4
                        // -> lanes 0..15 hit banks 0,4,...,60 (conflict-free)

// 128 threads = 4 wave32. Each wave: 16 n-rows x full D=64 (4 C-tiles of
// 16x16, K=64 -> 16 WMMA each). Block: 64 n-rows of one part p.
__global__ __launch_bounds__(128)
void sepfc_wmma_f32(const float* __restrict__ x,
                    const float* __restrict__ W,
                    const float* __restrict__ bias,
                    float* __restrict__ out) {
    __shared__ float lds_w[D_DIM * LDS_STRIDE];   // ~17 KB, W_p row-major [d][c] padded

    const int p    = blockIdx.y;
    const int tid  = threadIdx.x;
    const int wave = tid >> 5;          // 0..3
    const int lane = tid & 31;

    // ---- cooperative stage of W_p (64x64 fp32 = 16KB) into padded LDS ----
    const float* __restrict__ wp = W + (size_t)p * D_DIM * D_DIM;
    #pragma unroll
    for (int i = tid; i < D_DIM * (D_DIM / 2); i += 128) {   // 2048 float2 chunks
        const int r  = i >> 5;          // row d: 0..63
        const int cc = (i & 31) * 2;    // col c: 0,2,...,62 (8B aligned both sides)
        *(float2*)(lds_w + r * LDS_STRIDE + cc) =
            *(const float2*)(wp + r * D_DIM + cc);
    }
    __syncthreads();

    // ---- A fragments: x tile 16(n) x 64(k), loaded once, reused 4x -------
    // A 16x4 fp32 layout: lanes 0-15 -> M=lane, VGPR v -> K=v; lanes 16-31 -> K=2+v.
    const int row   = lane & 15;
    const int khalf = lane >> 4;                      // 0 or 1
    const int n0    = blockIdx.x * 64 + wave * 16;    // wave's n base

    const float* __restrict__ xrow =
        x + ((size_t)(n0 + row) * P_TOT + p) * D_DIM;

    v2f a[16];
    #pragma unroll
    for (int t = 0; t < 16; ++t) {
        a[t] = *(const v2f*)(xrow + 4 * t + 2 * khalf);   // k = 4t + 2*khalf + {0,1}
    }

    float* __restrict__ obase = out + ((size_t)n0 * P_TOT + p) * D_DIM;
    const float* __restrict__ bp = bias + (size_t)p * D_DIM;

    // ---- 4 output tiles of 16x16 along d ---------------------------------
    #pragma unroll
    for (int dt = 0; dt < 4; ++dt) {
        const int d = dt * 16 + (lane & 15);          // this lane's N column

        // C/D layout: lane N = lane&15, VGPR v -> M = v + 8*khalf.
        // Bias depends only on N -> broadcast into all 8 accumulator regs.
        const float bb = bp[d];
        v8f c = {bb, bb, bb, bb, bb, bb, bb, bb};

        #pragma unroll
        for (int t = 0; t < 16; ++t) {
            // B 4x16 fp32: lane N = lane&15 -> column d; VGPR v -> K = 4t+2*khalf+v.
            // B[k][n] = W[p, n, k]  (out = x @ W^T), so read W_p row d, cols k.
            v2f bfrag = *(const v2f*)(lds_w + d * LDS_STRIDE + 4 * t + 2 * khalf);
            c = __builtin_amdgcn_wmma_f32_16x16x4_f32(
                    /*neg_a=*/false, a[t],
                    /*neg_b=*/false, bfrag,
                    /*c_mod=*/(short)0, c,
                    /*reuse_a=*/false, /*reuse_b=*/false);
        }

        // Store: VGPR v holds (M = v + 8*khalf, N = d). Each store instr writes
        // two 64B runs (lanes 0-15 row m, lanes 16-31 row m+8) -> coalesced.
        #pragma unroll
        for (int v = 0; v < 8; ++v) {
            const int m = v + 8 * khalf;
            obase[(size_t)m * P_TOT * D_DIM + d] = c[v];
        }
    }
}

extern "C" void kernel_launch(void* const* d_in, const int* in_sizes, int n_in,
                              void* d_out, int out_size, void* d_ws, size_t ws_size,
                              hipStream_t stream) {
    (void)in_sizes; (void)n_in; (void)out_size; (void)d_ws; (void)ws_size;
    const float* x = (const float*)d_in[0];
    const float* W = (const float*)d_in[1];
    const float* b = (const float*)d_in[2];
    float* out     = (float*)d_out;

    dim3 grid(N_TOT / 64, P_TOT);   // 128 x 128 blocks
    sepfc_wmma_f32<<<grid, 128, 0, stream>>>(x, W, b, out);
}